// Loss_yolov1_5325759447816
// MI455X (gfx1250) — compile-verified
//
#include <hip/hip_runtime.h>
#include <stdint.h>
#include <stddef.h>

#define G 7
#define CELLS 49                 // G*G
#define CH 30
#define ITEM_F (CH * CELLS)      // 1470 floats per (pred|labels) per batch item
#define ITEMS_PER_BLOCK 4        // 4*1470 floats = 23520 B per array -> 1470 exact b128 units
#define BLOCK 256

// ---- CDNA5 async global->LDS path (guarded; falls back to direct loads) ----
#if defined(__gfx1250__) && \
    __has_builtin(__builtin_amdgcn_global_load_async_to_lds_b128) && \
    __has_builtin(__builtin_amdgcn_global_load_async_to_lds_b32)
#define USE_ASYNC 1
#else
#define USE_ASYNC 0
#endif

#if USE_ASYNC
typedef int v4i __attribute__((vector_size(16)));
typedef __attribute__((address_space(1))) v4i* g128p;   // global, int4 pointee
typedef __attribute__((address_space(3))) v4i* l128p;   // LDS,    int4 pointee
typedef __attribute__((address_space(1))) int* g32p;    // global, int pointee
typedef __attribute__((address_space(3))) int* l32p;    // LDS,    int pointee

__device__ __forceinline__ void wait_async_zero() {
#if __has_builtin(__builtin_amdgcn_s_wait_asynccnt)
  __builtin_amdgcn_s_wait_asynccnt(0);
#else
  asm volatile("s_wait_asynccnt 0" ::: "memory");
#endif
}
#endif

// ----------------------------- per-cell math --------------------------------
__device__ __forceinline__ float iou_fn(float x1a, float y1a, float x2a, float y2a,
                                        float x1b, float y1b, float x2b, float y2b) {
  float iw = fmaxf(fminf(x2a, x2b) - fmaxf(x1a, x1b), 0.0f);
  float ih = fmaxf(fminf(y2a, y2b) - fmaxf(y1a, y1b), 0.0f);
  float inter = iw * ih;
  float a1 = (x2a - x1a) * (y2a - y1a);
  float a2 = (x2b - x1b) * (y2b - y1b);
  float uni = a1 + a2 - inter;
  return inter > 0.0f ? inter / uni : 0.0f;
}

// p,l point at channel-0 of this cell; channel c lives at [c*CELLS].
__device__ __forceinline__ float cell_loss(const float* __restrict__ p,
                                           const float* __restrict__ l,
                                           float fm, float fn) {
  const float invG = 1.0f / (float)G;
  float p0 = p[0*CELLS], p1 = p[1*CELLS], p2 = p[2*CELLS], p3 = p[3*CELLS], p4 = p[4*CELLS];
  float p5 = p[5*CELLS], p6 = p[6*CELLS], p7 = p[7*CELLS], p8 = p[8*CELLS], p9 = p[9*CELLS];
  float l0 = l[0*CELLS], l1 = l[1*CELLS], l2 = l[2*CELLS], l3 = l[3*CELLS], l4 = l[4*CELLS];
  float l5 = l[5*CELLS], l6 = l[6*CELLS], l7 = l[7*CELLS], l8 = l[8*CELLS];

  // box1 (pred 0..3), box2 (pred 5..8), ground truth (labels 0..3)
  float cx1 = (p0 + fm) * invG, cy1 = (p1 + fn) * invG;
  float b1x1 = cx1 - 0.5f*p2, b1y1 = cy1 - 0.5f*p3, b1x2 = cx1 + 0.5f*p2, b1y2 = cy1 + 0.5f*p3;
  float cx2 = (p5 + fm) * invG, cy2 = (p6 + fn) * invG;
  float b2x1 = cx2 - 0.5f*p7, b2y1 = cy2 - 0.5f*p8, b2x2 = cx2 + 0.5f*p7, b2y2 = cy2 + 0.5f*p8;
  float cxg = (l0 + fm) * invG, cyg = (l1 + fn) * invG;
  float bgx1 = cxg - 0.5f*l2, bgy1 = cyg - 0.5f*l3, bgx2 = cxg + 0.5f*l2, bgy2 = cyg + 0.5f*l3;

  float iou1 = iou_fn(b1x1, b1y1, b1x2, b1y2, bgx1, bgy1, bgx2, bgy2);
  float iou2 = iou_fn(b2x1, b2y1, b2x2, b2y2, bgx1, bgy1, bgx2, bgy2);

  float d0 = p0 - l0, d1 = p1 - l1;
  float s2 = sqrtf(p2) - sqrtf(l2), s3 = sqrtf(p3) - sqrtf(l3);
  float coor1 = 5.0f * (d0*d0 + d1*d1 + s2*s2 + s3*s3);
  float d5 = p5 - l5, d6 = p6 - l6;
  float s7 = sqrtf(p7) - sqrtf(l7), s8 = sqrtf(p8) - sqrtf(l8);
  float coor2 = 5.0f * (d5*d5 + d6*d6 + s7*s7 + s8*s8);

  float e1 = p4 - iou1, e2 = p9 - iou2;
  float loss_b1 = coor1 + e1*e1 + 0.5f*e2*e2;
  float loss_b2 = coor2 + e2*e2 + 0.5f*e1*e1;

  float cls = 0.0f;
#pragma unroll
  for (int c = 10; c < CH; ++c) {
    float d = p[c*CELLS] - l[c*CELLS];
    cls += d * d;
  }

  float obj_loss   = (iou1 >= iou2 ? loss_b1 : loss_b2) + cls;
  float noobj_loss = 0.5f * (p4*p4 + p9*p9);
  return (l4 == 1.0f) ? obj_loss : noobj_loss;
}

// ------------------------------- kernel 1 -----------------------------------
__global__ __launch_bounds__(BLOCK) void yolo_partial(
    const float* __restrict__ pred, const float* __restrict__ labels,
    float* __restrict__ partials, int B) {
  int b0 = blockIdx.x * ITEMS_PER_BLOCK;
  int items = B - b0;
  if (items > ITEMS_PER_BLOCK) items = ITEMS_PER_BLOCK;
  int nf = items * ITEM_F;  // floats per array staged by this block

#if USE_ASYNC
  __shared__ __align__(16) float smem[2 * ITEM_F * ITEMS_PER_BLOCK];  // 47040 B
  float* sp = smem;
  float* sl = smem + ITEM_F * ITEMS_PER_BLOCK;
  {
    const float* gp = pred   + (size_t)b0 * ITEM_F;
    const float* gl = labels + (size_t)b0 * ITEM_F;
    int n128 = nf >> 2;   // 16-byte units
    int rem  = nf & 3;
    for (int u = threadIdx.x; u < n128; u += BLOCK) {
      __builtin_amdgcn_global_load_async_to_lds_b128((g128p)(gp + 4*u), (l128p)(sp + 4*u), 0, 0);
      __builtin_amdgcn_global_load_async_to_lds_b128((g128p)(gl + 4*u), (l128p)(sl + 4*u), 0, 0);
    }
    if ((int)threadIdx.x < rem) {
      int u = (n128 << 2) + threadIdx.x;
      __builtin_amdgcn_global_load_async_to_lds_b32((g32p)(gp + u), (l32p)(sp + u), 0, 0);
      __builtin_amdgcn_global_load_async_to_lds_b32((g32p)(gl + u), (l32p)(sl + u), 0, 0);
    }
    wait_async_zero();   // my wave's copies done
    __syncthreads();     // everyone's copies done
  }
#endif

  float acc = 0.0f;
  int ncells = items * CELLS;
  for (int t = threadIdx.x; t < ncells; t += BLOCK) {
    int item = t / CELLS;
    int cell = t - item * CELLS;
#if USE_ASYNC
    const float* p = sp + item * ITEM_F + cell;   // LDS, conflict-free stride-49 reads
    const float* l = sl + item * ITEM_F + cell;
#else
    const float* p = pred   + (size_t)(b0 + item) * ITEM_F + cell;
    const float* l = labels + (size_t)(b0 + item) * ITEM_F + cell;
#endif
    acc += cell_loss(p, l, (float)(cell / G), (float)(cell % G));
  }

  // wave32 reduce, then cross-wave via LDS
#pragma unroll
  for (int off = 16; off > 0; off >>= 1) acc += __shfl_down(acc, off, 32);
  __shared__ float wsum[BLOCK / 32];
  if ((threadIdx.x & 31) == 0) wsum[threadIdx.x >> 5] = acc;
  __syncthreads();
  if (threadIdx.x < (BLOCK / 32)) {
    float v = wsum[threadIdx.x];
#pragma unroll
    for (int off = BLOCK / 64; off > 0; off >>= 1) v += __shfl_down(v, off, 32);
    if (threadIdx.x == 0) partials[blockIdx.x] = v;
  }
}

// ------------------------------- kernel 2 -----------------------------------
__global__ __launch_bounds__(1024) void yolo_final(
    const float* __restrict__ partials, int n, float inv_b, float* __restrict__ out) {
  float acc = 0.0f;
  for (int i = threadIdx.x; i < n; i += blockDim.x) acc += partials[i];
#pragma unroll
  for (int off = 16; off > 0; off >>= 1) acc += __shfl_down(acc, off, 32);
  __shared__ float w[32];
  if ((threadIdx.x & 31) == 0) w[threadIdx.x >> 5] = acc;
  __syncthreads();
  if (threadIdx.x == 0) {
    float s = 0.0f;
    int nw = (int)(blockDim.x >> 5);
    for (int i = 0; i < nw; ++i) s += w[i];
    out[0] = s * inv_b;   // overwrite: deterministic, no atomics
  }
}

// ------------------------------- launcher -----------------------------------
extern "C" void kernel_launch(void* const* d_in, const int* in_sizes, int n_in,
                              void* d_out, int out_size, void* d_ws, size_t ws_size,
                              hipStream_t stream) {
  const float* pred   = (const float*)d_in[0];
  const float* labels = (const float*)d_in[1];
  int B = in_sizes[0] / ITEM_F;                       // 32768 for the reference setup
  int nblocks = (B + ITEMS_PER_BLOCK - 1) / ITEMS_PER_BLOCK;
  float* partials = (float*)d_ws;                     // nblocks * 4 bytes scratch

  yolo_partial<<<nblocks, BLOCK, 0, stream>>>(pred, labels, partials, B);
  yolo_final<<<1, 1024, 0, stream>>>(partials, nblocks, 1.0f / (float)B, (float*)d_out);
}